// NeuralGNNWithAttention_1331439862293
// MI455X (gfx1250) — compile-verified
//
#include <hip/hip_runtime.h>
#include <hip/hip_bf16.h>

typedef float v2f __attribute__((ext_vector_type(2)));
typedef float v8f __attribute__((ext_vector_type(8)));

__device__ __forceinline__ v8f wmma_f32x4(v2f a, v2f b, v8f c) {
    // D(16x16,f32) = A(16x4,f32) * B(4x16,f32) + C
    return __builtin_amdgcn_wmma_f32_16x16x4_f32(false, a, false, b, (short)0, c, false, false);
}

// ---------------------------------------------------------------------------
// WMMA GEMM: C[M,Nn] = A[M,K] @ (TRANSB ? B[Nn,K]^T : B[K,Nn]) (+bias)(+relu)
// Each wave computes a 16(M) x 64(N) strip: the A fragment is loaded once per
// k-step and fed to 4 independent WMMAs (amortizes A traffic, deepens the XDL
// pipeline). M % 16 == 0, K % 4 == 0 (true here: K in {128,64,32}); Nn arbitrary.
// ---------------------------------------------------------------------------
template<int TRANSB, int BIAS, int RELU>
__global__ void gemm_wmma_kernel(const float* __restrict__ A, const float* __restrict__ B,
                                 const float* __restrict__ bias, float* __restrict__ C,
                                 int M, int Nn, int K) {
    int wave = threadIdx.x >> 5;
    int lane = threadIdx.x & 31;
    int half = lane >> 4;       // 0: lanes 0-15, 1: lanes 16-31
    int lm   = lane & 15;
    int tilesN4 = (Nn + 63) >> 6;           // 64-wide column groups
    int tilesM  = M >> 4;
    int t = blockIdx.x * (blockDim.x >> 5) + wave;
    if (t >= tilesM * tilesN4) return;       // wave-uniform exit (EXEC stays all-1)
    int tm = t / tilesN4, tn = t - tm * tilesN4;
    int row  = tm * 16 + lm;
    int col0 = tn * 64 + lm;                 // column for sub-tile j is col0 + 16*j
    v8f acc[4];
#pragma unroll
    for (int j = 0; j < 4; j++) acc[j] = (v8f){0.f,0.f,0.f,0.f,0.f,0.f,0.f,0.f};
    const float* arow = A + (size_t)row * K;
    for (int k0 = 0; k0 < K; k0 += 4) {
        int ka = k0 + 2 * half;
        v2f a; a.x = arow[ka]; a.y = arow[ka + 1];
#pragma unroll
        for (int j = 0; j < 4; j++) {
            int col = col0 + 16 * j;
            bool ok = col < Nn;
            v2f b;
            if (TRANSB) {
                const float* bp = B + (size_t)col * K + ka;
                b.x = ok ? bp[0] : 0.f;
                b.y = ok ? bp[1] : 0.f;
            } else {
                b.x = ok ? B[(size_t)ka * Nn + col] : 0.f;
                b.y = ok ? B[(size_t)(ka + 1) * Nn + col] : 0.f;
            }
            acc[j] = wmma_f32x4(a, b, acc[j]);
        }
    }
#pragma unroll
    for (int j = 0; j < 4; j++) {
        int col = col0 + 16 * j;
        if (col < Nn) {
            float bv = BIAS ? bias[col] : 0.f;
#pragma unroll
            for (int i = 0; i < 8; i++) {
                int r = tm * 16 + i + 8 * half;
                float v = acc[j][i] + bv;
                if (RELU) v = fmaxf(v, 0.f);
                C[(size_t)r * Nn + col] = v;
            }
        }
    }
}

// ---------------------------------------------------------------------------
// GAT helpers
// ---------------------------------------------------------------------------
__global__ void fill_f32_kernel(float* p, float v, int n) {
    int i = blockIdx.x * blockDim.x + threadIdx.x;
    if (i < n) p[i] = v;
}
__global__ void fill_u32_kernel(unsigned* p, unsigned v, int n) {
    int i = blockIdx.x * blockDim.x + threadIdx.x;
    if (i < n) p[i] = v;
}

// a_s[n,h] = sum_c h[n,h,c]*att_src[h,c];  a_d likewise
__global__ void gat_scores_kernel(const float* __restrict__ hbuf,
                                  const float* __restrict__ att_s,
                                  const float* __restrict__ att_d,
                                  float* __restrict__ asb, float* __restrict__ adb,
                                  int n, int H) {
    int idx = blockIdx.x * blockDim.x + threadIdx.x;
    if (idx >= n * H) return;
    int hh = idx % H;
    const float* hp = hbuf + (size_t)idx * 64;
    const float* ap = att_s + hh * 64;
    const float* dp = att_d + hh * 64;
    float sa = 0.f, sd = 0.f;
#pragma unroll 8
    for (int c = 0; c < 64; c++) { float v = hp[c]; sa += v * ap[c]; sd += v * dp[c]; }
    asb[idx] = sa; adb[idx] = sd;
}

__device__ __forceinline__ unsigned enc_f32(float v) {
    unsigned u = __float_as_uint(v);
    return (u & 0x80000000u) ? ~u : (u | 0x80000000u);
}
__device__ __forceinline__ float dec_f32(unsigned u) {
    unsigned b = (u & 0x80000000u) ? (u ^ 0x80000000u) : ~u;
    return __uint_as_float(b);
}

// pass 1: e = leaky_relu(a_s[src]+a_d[dst]); segment max over dst (monotone-uint atomicMax)
__global__ void edge_max_kernel(const int* __restrict__ srcp, const int* __restrict__ dstp,
                                const float* __restrict__ asb, const float* __restrict__ adb,
                                unsigned* __restrict__ menc, float* __restrict__ ebuf,
                                int E, int Etot, int H) {
    int idx = blockIdx.x * blockDim.x + threadIdx.x;
    if (idx >= Etot * H) return;
    int e = idx / H, hh = idx - e * H;
    int s = e < E ? srcp[e] : (e - E);
    int d = e < E ? dstp[e] : (e - E);
    float v = asb[s * H + hh] + adb[d * H + hh];
    v = v >= 0.f ? v : 0.2f * v;           // leaky_relu(., 0.2)
    ebuf[idx] = v;
    atomicMax(&menc[d * H + hh], enc_f32(v));
}

// pass 2: ex = exp(e - m[dst]); segment sum over dst
__global__ void edge_exp_kernel(const int* __restrict__ dstp,
                                const unsigned* __restrict__ menc,
                                float* __restrict__ ebuf, float* __restrict__ sbuf,
                                int E, int Etot, int H) {
    int idx = blockIdx.x * blockDim.x + threadIdx.x;
    if (idx >= Etot * H) return;
    int e = idx / H, hh = idx - e * H;
    int d = e < E ? dstp[e] : (e - E);
    float m = dec_f32(menc[d * H + hh]);
    float ex = __expf(ebuf[idx] - m);
    ebuf[idx] = ex;
    atomicAdd(&sbuf[d * H + hh], ex);
}

// pass 3: out[dst,h,:] += h[src,h,:] * alpha ; one thread per (edge,h,4-channel group)
__global__ void edge_agg_kernel(const int* __restrict__ srcp, const int* __restrict__ dstp,
                                const float* __restrict__ hbuf, const float* __restrict__ ebuf,
                                const float* __restrict__ sbuf, float* __restrict__ agg,
                                int E, int Etot, int H) {
    int idx = blockIdx.x * blockDim.x + threadIdx.x;
    if (idx >= Etot * H * 16) return;
    int cg = idx & 15;
    int t  = idx >> 4;
    int e = t / H, hh = t - e * H;
    int s = e < E ? srcp[e] : (e - E);
    int d = e < E ? dstp[e] : (e - E);
    float alpha = ebuf[t] / (sbuf[d * H + hh] + 1e-16f);
    const float4 hv = *((const float4*)(hbuf + (size_t)(s * H + hh) * 64) + cg);
    float* op = agg + (size_t)(d * H + hh) * 64 + cg * 4;
    atomicAdd(op + 0, hv.x * alpha);
    atomicAdd(op + 1, hv.y * alpha);
    atomicAdd(op + 2, hv.z * alpha);
    atomicAdd(op + 3, hv.w * alpha);
}

// mean over heads + bias
__global__ void mean_heads_kernel(const float* __restrict__ agg, const float* __restrict__ bias,
                                  float* __restrict__ xout, int n, int H) {
    int idx = blockIdx.x * blockDim.x + threadIdx.x;
    if (idx >= n * 64) return;
    int node = idx >> 6, c = idx & 63;
    float s = 0.f;
    for (int hh = 0; hh < H; hh++) s += agg[(size_t)(node * H + hh) * 64 + c];
    xout[idx] = s / (float)H + bias[c];
}

// ---------------------------------------------------------------------------
// BatchNorm (biased variance, as reference) + ReLU
// ---------------------------------------------------------------------------
__global__ void bn_stats_kernel(const float* __restrict__ x, float* __restrict__ stats, int n) {
    __shared__ float sh[256], sh2[256];
    int c = blockIdx.x;
    float s = 0.f, s2 = 0.f;
    for (int r = threadIdx.x; r < n; r += blockDim.x) {
        float v = x[(size_t)r * 64 + c];
        s += v; s2 += v * v;
    }
    sh[threadIdx.x] = s; sh2[threadIdx.x] = s2;
    __syncthreads();
    for (int off = 128; off > 0; off >>= 1) {
        if (threadIdx.x < off) { sh[threadIdx.x] += sh[threadIdx.x + off]; sh2[threadIdx.x] += sh2[threadIdx.x + off]; }
        __syncthreads();
    }
    if (threadIdx.x == 0) {
        float mu = sh[0] / (float)n;
        stats[c] = mu;
        stats[64 + c] = sh2[0] / (float)n - mu * mu;
    }
}

__global__ void bn_apply_kernel(float* __restrict__ x, const float* __restrict__ stats,
                                const float* __restrict__ gamma, const float* __restrict__ beta,
                                int total) {
    int i = blockIdx.x * blockDim.x + threadIdx.x;
    if (i >= total) return;
    int c = i & 63;
    float v = (x[i] - stats[c]) * rsqrtf(stats[64 + c] + 1e-5f) * gamma[c] + beta[c];
    x[i] = fmaxf(v, 0.f);
}

// ---------------------------------------------------------------------------
// Flash attention (4 heads, hd=16), transposed so the softmax reduction axis
// stays inside each lane's registers:
//   S^T (16key x 16query) = K_tile @ (Q/4)_tile^T  (4 WMMAs; 1/sqrt(hd) folded
//                                                   into the Q fragment)
//   row reduce over keys  = in-lane max3/add chain + ONE shfl_xor(16)
//   O^T (16hd x 16query) += V_tile^T @ P^T         (4 WMMAs; A-frag direct from
//                                                   global, B-frag via LDS)
// The LDS P^T tile is private to the wave and CDNA5 keeps same-wave LDS ops in
// order (DScnt), so only a compiler re-ordering fence (wave_barrier) is needed
// -- no workgroup s_barrier in the inner loop.
// ---------------------------------------------------------------------------
__global__ void flash_attn_kernel(const float* __restrict__ Q, const float* __restrict__ K,
                                  const float* __restrict__ V, float* __restrict__ O, int n) {
    __shared__ float pl[4][16][17];   // [wave][key][query], padded
    int wave = threadIdx.x >> 5;
    int lane = threadIdx.x & 31;
    int half = lane >> 4;
    int lm   = lane & 15;
    int qtiles = n >> 4;
    int gw = blockIdx.x * (blockDim.x >> 5) + wave;   // grid sized exactly: 4*qtiles waves
    int h  = gw / qtiles;
    int qt = gw - h * qtiles;

    // B-fragment of Q^T (col = query = lm), loaded once, pre-scaled by 1/sqrt(hd)
    v2f qfrag[4];
#pragma unroll
    for (int s = 0; s < 4; s++) {
        const float* qp = Q + (size_t)(qt * 16 + lm) * 64 + h * 16 + s * 4 + 2 * half;
        qfrag[s].x = qp[0] * 0.25f;
        qfrag[s].y = qp[1] * 0.25f;
    }
    float m_run = -1e30f, l_run = 0.f;    // per query (lane%16), replicated per half
    v8f o = {0.f,0.f,0.f,0.f,0.f,0.f,0.f,0.f};   // o[i] = O^T[hd = i+8*half][query = lm]

    for (int kt = 0; kt < qtiles; kt++) {
        // S^T = K @ (Q/4)^T : A-frag rows are keys
        v8f sa = {0.f,0.f,0.f,0.f,0.f,0.f,0.f,0.f};
#pragma unroll
        for (int s = 0; s < 4; s++) {
            const float* kp = K + (size_t)(kt * 16 + lm) * 64 + h * 16 + s * 4 + 2 * half;
            v2f a; a.x = kp[0]; a.y = kp[1];
            sa = wmma_f32x4(a, qfrag[s], sa);
        }
        // sa[i] = S^T[key = i+8*half][query = lm]
        float mx = sa[0];
#pragma unroll
        for (int i = 1; i < 8; i++) mx = fmaxf(mx, sa[i]);
        mx = fmaxf(mx, __shfl_xor(mx, 16, 32));          // combine the two key halves
        float mnew = fmaxf(m_run, mx);
        float p[8]; float su = 0.f;
#pragma unroll
        for (int i = 0; i < 8; i++) { p[i] = __expf(sa[i] - mnew); su += p[i]; }
        su += __shfl_xor(su, 16, 32);
        float sc = __expf(m_run - mnew);
        l_run = l_run * sc + su;
        m_run = mnew;
#pragma unroll
        for (int i = 0; i < 8; i++) o[i] *= sc;
        // stage P^T for the B-fragment (wave-private LDS slice; same-wave LDS is
        // in-order on CDNA5, so only a compiler fence is required)
#pragma unroll
        for (int i = 0; i < 8; i++) pl[wave][i + 8 * half][lm] = p[i];
        __builtin_amdgcn_wave_barrier();
        // O^T += V^T @ P^T : A-frag is V^T (rows = hd = lm), straight from global
#pragma unroll
        for (int s = 0; s < 4; s++) {
            int kk = s * 4 + 2 * half;
            const float* vp = V + (size_t)(kt * 16 + kk) * 64 + h * 16 + lm;
            v2f a; a.x = vp[0]; a.y = vp[64];
            v2f b; b.x = pl[wave][kk][lm]; b.y = pl[wave][kk + 1][lm];
            o = wmma_f32x4(a, b, o);
        }
        __builtin_amdgcn_wave_barrier();
    }
    // epilogue: O[query][hd] ; per lane hd = 8*half .. 8*half+7 is contiguous
    float inv = 1.f / l_run;
    float* op = O + (size_t)(qt * 16 + lm) * 64 + h * 16 + 8 * half;
    float4 r0 = {o[0] * inv, o[1] * inv, o[2] * inv, o[3] * inv};
    float4 r1 = {o[4] * inv, o[5] * inv, o[6] * inv, o[7] * inv};
    *((float4*)op)     = r0;
    *((float4*)op + 1) = r1;
}

// ---------------------------------------------------------------------------
// Orchestration
// ---------------------------------------------------------------------------
static inline int cdiv_i(int a, int b) { return (a + b - 1) / b; }

extern "C" void kernel_launch(void* const* d_in, const int* in_sizes, int n_in,
                              void* d_out, int out_size, void* d_ws, size_t ws_size,
                              hipStream_t stream) {
    const float* x   = (const float*)d_in[0];
    const int*   ei  = (const int*)d_in[1];
    const float* W[3]  = {(const float*)d_in[2], (const float*)d_in[6],  (const float*)d_in[10]};
    const float* As[3] = {(const float*)d_in[3], (const float*)d_in[7],  (const float*)d_in[11]};
    const float* Ad[3] = {(const float*)d_in[4], (const float*)d_in[8],  (const float*)d_in[12]};
    const float* Bs[3] = {(const float*)d_in[5], (const float*)d_in[9],  (const float*)d_in[13]};
    const float* bn_g = (const float*)d_in[14];
    const float* bn_b = (const float*)d_in[15];
    const float* Wq = (const float*)d_in[16];
    const float* Wk = (const float*)d_in[17];
    const float* Wv = (const float*)d_in[18];
    const float* Wo = (const float*)d_in[19];
    const float* bq = (const float*)d_in[20];
    const float* bk = (const float*)d_in[21];
    const float* bv = (const float*)d_in[22];
    const float* bo = (const float*)d_in[23];
    const float* Wc1 = (const float*)d_in[24];
    const float* bc1 = (const float*)d_in[25];
    const float* Wc2 = (const float*)d_in[26];
    const float* bc2 = (const float*)d_in[27];

    const int N = 8192, C = 64;
    const int E = in_sizes[1] / 2;
    const int Etot = E + N;

    float* w = (float*)d_ws;
    float* buf_h = w;                  w += (size_t)N * 512;       // h = x@W  (max N x 512)
    float* agg   = w;                  w += (size_t)N * 512;       // segment-sum accumulator
    float* ebuf  = w;                  w += (size_t)Etot * 8;      // per-edge e / exp values
    float* asb   = w;                  w += (size_t)N * 8;
    float* adb   = w;                  w += (size_t)N * 8;
    float* sbuf  = w;                  w += (size_t)N * 8;
    unsigned* menc = (unsigned*)w;     w += (size_t)N * 8;
    float* xcur  = w;                  w += (size_t)N * 64;
    float* xtmp  = w;                  w += (size_t)N * 64;
    float* qb    = w;                  w += (size_t)N * 64;
    float* kb    = w;                  w += (size_t)N * 64;
    float* vb    = w;                  w += (size_t)N * 64;
    float* atb   = w;                  w += (size_t)N * 64;
    float* m1    = w;                  w += (size_t)N * 32;
    float* stats = w;                  w += 128;

    auto gemm_blocks = [](int M, int Nn) { return ((M / 16) * ((Nn + 63) / 64) + 7) / 8; };

    const int heads[3] = {8, 8, 1};
    const float* cur = x;
    int fin = 128;
    for (int L = 0; L < 3; L++) {
        int H = heads[L];
        int outc = H * C;
        // h = x @ W_L  (WMMA)
        gemm_wmma_kernel<0,0,0><<<gemm_blocks(N, outc), 256, 0, stream>>>(cur, W[L], nullptr, buf_h, N, outc, fin);
        // attention logits per node
        gat_scores_kernel<<<cdiv_i(N * H, 256), 256, 0, stream>>>(buf_h, As[L], Ad[L], asb, adb, N, H);
        // reset accumulators
        fill_u32_kernel<<<cdiv_i(N * H, 256), 256, 0, stream>>>(menc, 0u, N * H);
        fill_f32_kernel<<<cdiv_i(N * H, 256), 256, 0, stream>>>(sbuf, 0.f, N * H);
        fill_f32_kernel<<<cdiv_i(N * H * C, 256), 256, 0, stream>>>(agg, 0.f, N * H * C);
        // segment softmax + weighted aggregation over edges (+self loops)
        int eh = Etot * H;
        edge_max_kernel<<<cdiv_i(eh, 256), 256, 0, stream>>>(ei, ei + E, asb, adb, menc, ebuf, E, Etot, H);
        edge_exp_kernel<<<cdiv_i(eh, 256), 256, 0, stream>>>(ei + E, menc, ebuf, sbuf, E, Etot, H);
        edge_agg_kernel<<<cdiv_i(eh * 16, 256), 256, 0, stream>>>(ei, ei + E, buf_h, ebuf, sbuf, agg, E, Etot, H);
        // mean over heads + bias, then BN + ReLU
        mean_heads_kernel<<<cdiv_i(N * C, 256), 256, 0, stream>>>(agg, Bs[L], xcur, N, H);
        bn_stats_kernel<<<64, 256, 0, stream>>>(xcur, stats, N);
        bn_apply_kernel<<<cdiv_i(N * C, 256), 256, 0, stream>>>(xcur, stats, bn_g + L * 64, bn_b + L * 64, N * C);
        cur = xcur; fin = C;
    }

    // MHA: projections (WMMA, transB + bias), flash attention (WMMA), output proj
    gemm_wmma_kernel<1,1,0><<<gemm_blocks(N, 64), 256, 0, stream>>>(xcur, Wq, bq, qb, N, 64, 64);
    gemm_wmma_kernel<1,1,0><<<gemm_blocks(N, 64), 256, 0, stream>>>(xcur, Wk, bk, kb, N, 64, 64);
    gemm_wmma_kernel<1,1,0><<<gemm_blocks(N, 64), 256, 0, stream>>>(xcur, Wv, bv, vb, N, 64, 64);
    flash_attn_kernel<<<N / 16, 128, 0, stream>>>(qb, kb, vb, atb, N);
    gemm_wmma_kernel<1,1,0><<<gemm_blocks(N, 64), 256, 0, stream>>>(atb, Wo, bo, xtmp, N, 64, 64);

    // classifier MLP
    gemm_wmma_kernel<1,1,1><<<gemm_blocks(N, 32), 256, 0, stream>>>(xtmp, Wc1, bc1, m1, N, 32, 64);
    gemm_wmma_kernel<1,1,0><<<gemm_blocks(N, 10), 256, 0, stream>>>(m1, Wc2, bc2, (float*)d_out, N, 10, 32);
}